// AttentionPruneKV_70918499992291
// MI455X (gfx1250) — compile-verified
//
#include <hip/hip_runtime.h>
#include <math.h>

#define B_    4
#define NH    8
#define N_    1024
#define DH    64
#define DIM   512
#define INNER 512
#define QKV3  1536
#define SCALE 0.125f          // 64^-0.5
#define ROWS  16
#define RSTR  1028            // padded LDS row stride (floats): 1028 % 64 == 4 -> conflict-free column reads
#define SIG_GATE 0.11920292202211755f  // sigmoid(-2.0)

typedef __attribute__((ext_vector_type(2))) float v2f;
typedef __attribute__((ext_vector_type(8))) float v8f;

static __device__ __forceinline__ v8f wmma4(v2f a, v2f b, v8f c) {
  // D = A(16x4,f32) * B(4x16,f32) + C(16x16,f32)
  return __builtin_amdgcn_wmma_f32_16x16x4_f32(false, a, false, b, (short)0, c, false, false);
}

// ---------------------------------------------------------------------------
// Kernel 1: qkv = x @ W_qkv  (4096x512 @ 512x1536), scatter into head-major q/k/v
// One 16x16 output tile per wave, 8 waves per block.
// ---------------------------------------------------------------------------
__global__ void __launch_bounds__(256) k_qkv(const float* __restrict__ x,
                                             const float* __restrict__ wqkv,
                                             float* __restrict__ qb,
                                             float* __restrict__ kb,
                                             float* __restrict__ vb) {
  const int lane = threadIdx.x & 31;
  const int wave = threadIdx.x >> 5;
  const int tile = blockIdx.x * 8 + wave;       // 256*96 = 24576 tiles
  const int rt = tile / 96;
  const int ct = tile % 96;
  const int hl = lane & 15, hh = lane >> 4;
  const int row = rt * 16 + hl;
  const int col = ct * 16 + hl;

  v8f c = {};
  for (int kk = 0; kk < DIM / 4; ++kk) {
    const int k0 = kk * 4 + 2 * hh;
    float2 t = *(const float2*)(x + (size_t)row * DIM + k0);
    v2f a; a[0] = t.x; a[1] = t.y;
    v2f b;
    b[0] = wqkv[(size_t)(k0 + 0) * QKV3 + col];
    b[1] = wqkv[(size_t)(k0 + 1) * QKV3 + col];
    c = wmma4(a, b, c);
  }

  // scatter: col -> (which, head, d); constant per lane (16 | 64 | 512 alignment)
  const int which = col / INNER;
  const int hcol  = col % INNER;
  const int head  = hcol / DH;
  const int d     = hcol % DH;
  float* dst = (which == 0) ? qb : ((which == 1) ? kb : vb);
  for (int r = 0; r < 8; ++r) {
    const int m    = r + 8 * hh;
    const int grow = rt * 16 + m;
    const int bb   = grow >> 10;
    const int nn   = grow & 1023;
    dst[(((size_t)bb * NH + head) * N_ + nn) * DH + d] = c[r];
  }
}

// ---------------------------------------------------------------------------
// Kernel 2: fused attention for one (b, h, 16-query block) per workgroup.
//   phase 1: dots0 / dots (bias + pos-embed fused) into LDS via WMMA
//   phase 2: dual softmax (attn0 streamed to output), dynamic-threshold prune
//   phase 3: pruned-attn @ V via WMMA from LDS
// ---------------------------------------------------------------------------
__global__ void __launch_bounds__(256, 1) k_attn(
    const float* __restrict__ qb, const float* __restrict__ kb,
    const float* __restrict__ vb, const float* __restrict__ prob,
    const float* __restrict__ table1, const float* __restrict__ table2,
    const float* __restrict__ headsita, const float* __restrict__ wthr,
    const int* __restrict__ relidx, const float* __restrict__ dis,
    float* __restrict__ attn0_out, float* __restrict__ ob) {
  extern __shared__ float sm[];
  float* sd0 = sm;                // [16][RSTR]: dots0 -> exp(dots0 - m0)
  float* sd  = sm + ROWS * RSTR;  // [16][RSTR]: dots  -> pruned attn

  const int lane  = threadIdx.x & 31;
  const int wave  = threadIdx.x >> 5;
  const int bh    = blockIdx.x >> 6;   // 0..31
  const int itile = blockIdx.x & 63;
  const int b     = bh >> 3;
  const int h     = bh & 7;
  const int ibase = itile * 16;
  const int hl = lane & 15, hh = lane >> 4;

  const float hs     = headsita[h];
  const float factor = 1.0f / (2.0f * hs * hs + 1e-6f);

  // Preload Q-block A-fragments (16 k-steps of 4)
  v2f afrag[16];
  {
    const float* qrow = qb + ((size_t)bh * N_ + ibase + hl) * DH;
    for (int kk = 0; kk < 16; ++kk) {
      float2 t = *(const float2*)(qrow + kk * 4 + 2 * hh);
      afrag[kk][0] = t.x; afrag[kk][1] = t.y;
    }
  }

  // ---- Phase 1: scores ----------------------------------------------------
  for (int jt = wave; jt < 64; jt += 8) {
    const int jbase = jt * 16;
    const float* krow = kb + ((size_t)bh * N_ + jbase + hl) * DH;
    v8f c = {};
    for (int kk = 0; kk < 16; ++kk) {
      float2 t = *(const float2*)(krow + kk * 4 + 2 * hh);  // B[k][j] = K[j][k]
      v2f bf; bf[0] = t.x; bf[1] = t.y;
      c = wmma4(afrag[kk], bf, c);
    }
    const int jcol = jbase + hl;
    for (int r = 0; r < 8; ++r) {
      const int m  = r + 8 * hh;
      const float d0 = c[r] * SCALE;
      const int gi = (ibase + m) * N_ + jcol;
      const int idx = relidx[gi];
      const float bias = table1[idx * NH + h] * table2[idx * NH + h];
      const float pe   = expf(-factor * dis[gi]);
      sd0[m * RSTR + jcol] = d0;
      sd [m * RSTR + jcol] = d0 + bias + 0.01f * pe;
    }
  }
  __syncthreads();

  // ---- Phase 2: softmax + dynamic-threshold pruning (2 rows per wave) -----
  for (int rr = 0; rr < 2; ++rr) {
    const int r = wave * 2 + rr;
    float* rowd0 = sd0 + r * RSTR;
    float* rowd  = sd  + r * RSTR;

    // thresh_raw = sigmoid(q . W_thresh) * sigmoid(-2)
    float2 qv = *(const float2*)(qb + ((size_t)bh * N_ + ibase + r) * DH + 2 * lane);
    float tpart = qv.x * wthr[2 * lane] + qv.y * wthr[2 * lane + 1];
    for (int o = 16; o > 0; o >>= 1) tpart += __shfl_xor(tpart, o, 32);
    const float traw = (1.0f / (1.0f + expf(-tpart))) * SIG_GATE;

    float m0 = -3.4e38f, md = -3.4e38f, mn = 3.4e38f;
    for (int j = lane; j < N_; j += 32) {
      m0 = fmaxf(m0, rowd0[j]);
      const float a1 = rowd[j];
      md = fmaxf(md, a1); mn = fminf(mn, a1);
    }
    for (int o = 16; o > 0; o >>= 1) {
      m0 = fmaxf(m0, __shfl_xor(m0, o, 32));
      md = fmaxf(md, __shfl_xor(md, o, 32));
      mn = fminf(mn, __shfl_xor(mn, o, 32));
    }
    float s0 = 0.f, s = 0.f;
    for (int j = lane; j < N_; j += 32) {
      const float e0 = expf(rowd0[j] - m0);
      const float e1 = expf(rowd[j]  - md);
      rowd0[j] = e0; rowd[j] = e1;
      s0 += e0; s += e1;
    }
    for (int o = 16; o > 0; o >>= 1) {
      s0 += __shfl_xor(s0, o, 32);
      s  += __shfl_xor(s,  o, 32);
    }
    const float invs0 = 1.0f / s0, invs = 1.0f / s;
    const float amax = invs;                  // exp(md-md)/s
    const float amin = expf(mn - md) * invs;
    const float thr  = amin + traw * (amax - amin);
    const float on   = (prob[b * N_ + ibase + r] < 0.9f) ? 1.0f : 0.0f;

    float deno = 0.f;
    for (int j = lane; j < N_; j += 32) {
      const float a = rowd[j] * invs;         // attn
      const float v = (a > thr) ? a * on : 0.0f;  // attn * record
      rowd[j] = v;
      deno += v;
    }
    for (int o = 16; o > 0; o >>= 1) deno += __shfl_xor(deno, o, 32);
    const float dn = 1.0f / (deno + 1e-6f);

    float* a0row = attn0_out + ((size_t)bh * N_ + ibase + r) * N_;
    for (int j = lane; j < N_; j += 32) {
      rowd[j] *= dn;                 // attn_pruned
      a0row[j] = rowd0[j] * invs0;   // attn0 = softmax(dots0)
    }
  }
  __syncthreads();

  // ---- Phase 3: out_head = attn_pruned(16x1024) @ V(1024x64) --------------
  if (wave < 4) {
    const int nt = wave;             // 16-wide d-column tile
    v8f c = {};
    for (int kk = 0; kk < N_ / 4; ++kk) {
      const int k0 = kk * 4 + 2 * hh;
      float2 t = *(const float2*)(sd + hl * RSTR + k0);   // A[m][k] from LDS
      v2f a; a[0] = t.x; a[1] = t.y;
      v2f bf;
      bf[0] = vb[((size_t)bh * N_ + k0 + 0) * DH + nt * 16 + hl];
      bf[1] = vb[((size_t)bh * N_ + k0 + 1) * DH + nt * 16 + hl];
      c = wmma4(a, bf, c);
    }
    for (int r = 0; r < 8; ++r) {
      const int m = r + 8 * hh;
      ob[((size_t)b * N_ + ibase + m) * INNER + h * DH + nt * 16 + hl] = c[r];
    }
  }
}

// ---------------------------------------------------------------------------
// Kernel 3: out = o @ W_out + b_out  (4096x512 @ 512x512)
// ---------------------------------------------------------------------------
__global__ void __launch_bounds__(256) k_proj(const float* __restrict__ ob,
                                              const float* __restrict__ wout,
                                              const float* __restrict__ bout,
                                              float* __restrict__ out) {
  const int lane = threadIdx.x & 31;
  const int wave = threadIdx.x >> 5;
  const int tile = blockIdx.x * 8 + wave;   // 256*32 = 8192 tiles
  const int rt = tile >> 5;
  const int ct = tile & 31;
  const int hl = lane & 15, hh = lane >> 4;
  const int row = rt * 16 + hl;
  const int col = ct * 16 + hl;

  v8f c = {};
  for (int kk = 0; kk < INNER / 4; ++kk) {
    const int k0 = kk * 4 + 2 * hh;
    float2 t = *(const float2*)(ob + (size_t)row * INNER + k0);
    v2f a; a[0] = t.x; a[1] = t.y;
    v2f b;
    b[0] = wout[(size_t)(k0 + 0) * INNER + col];
    b[1] = wout[(size_t)(k0 + 1) * INNER + col];
    c = wmma4(a, b, c);
  }
  const float bias = bout[col];
  for (int r = 0; r < 8; ++r) {
    const int m = r + 8 * hh;
    out[(size_t)(rt * 16 + m) * INNER + col] = c[r] + bias;
  }
}

// ---------------------------------------------------------------------------
extern "C" void kernel_launch(void* const* d_in, const int* in_sizes, int n_in,
                              void* d_out, int out_size, void* d_ws, size_t ws_size,
                              hipStream_t stream) {
  const float* x      = (const float*)d_in[0];
  const float* prob   = (const float*)d_in[1];
  const float* wqkv   = (const float*)d_in[2];
  const float* t1     = (const float*)d_in[3];
  const float* t2     = (const float*)d_in[4];
  const float* hsit   = (const float*)d_in[5];
  const float* wthr   = (const float*)d_in[6];
  const float* wout   = (const float*)d_in[7];
  const float* bout   = (const float*)d_in[8];
  const int*   relidx = (const int*)d_in[9];
  const float* dis    = (const float*)d_in[10];

  float* outp   = (float*)d_out;
  float* attn0p = outp + (size_t)B_ * N_ * INNER;   // second tuple element

  const size_t perBuf = (size_t)B_ * NH * N_ * DH;  // 2,097,152 floats
  float* qb = (float*)d_ws;
  float* kb = qb + perBuf;
  float* vb = kb + perBuf;
  float* ob = vb + perBuf;                           // (b, n, inner) layout

  const size_t ldsBytes = (size_t)2 * ROWS * RSTR * sizeof(float);  // 131,584 B
  (void)hipFuncSetAttribute(reinterpret_cast<const void*>(k_attn),
                            hipFuncAttributeMaxDynamicSharedMemorySize,
                            (int)ldsBytes);

  k_qkv <<<3072, 256, 0, stream>>>(x, wqkv, qb, kb, vb);
  k_attn<<<B_ * NH * (N_ / 16), 256, ldsBytes, stream>>>(
      qb, kb, vb, prob, t1, t2, hsit, wthr, relidx, dis, attn0p, ob);
  k_proj<<<1024, 256, 0, stream>>>(ob, wout, bout, outp);
}